// Shapelet_66932770341112
// MI455X (gfx1250) — compile-verified
//
#include <hip/hip_runtime.h>
#include <hip/hip_bf16.h>
#include <stdint.h>

namespace {
constexpr int kB = 16, kC = 8, kT = 512, kL = 24, kStride = 2, kN = 64, kM = 245;
constexpr int kWpad = 25;       // odd stride -> conflict-free LDS across 32 lanes
constexpr int kThreads = 512;   // 16 wave32 per block
}

typedef __attribute__((address_space(1))) int as1_int;  // global
typedef __attribute__((address_space(3))) int as3_int;  // LDS

#if __has_builtin(__builtin_amdgcn_global_load_async_to_lds_b32)
#define HAVE_ASYNC_BUILTIN 1
#endif

__device__ __forceinline__ void async_cp_b32(const void* g, void* l) {
#if defined(HAVE_ASYNC_BUILTIN)
  __builtin_amdgcn_global_load_async_to_lds_b32(
      (as1_int*)(uintptr_t)g,
      (as3_int*)(uint32_t)(uintptr_t)l, 0, 0);
#else
  asm volatile("global_load_async_to_lds_b32 %0, %1, off"
               :: "v"((uint32_t)(uintptr_t)l), "v"(g) : "memory");
#endif
}

__device__ __forceinline__ void wait_async0() {
#if __has_builtin(__builtin_amdgcn_s_wait_asynccnt)
  __builtin_amdgcn_s_wait_asynccnt(0);
#else
  asm volatile("s_wait_asynccnt 0" ::: "memory");
#endif
}

__global__ __launch_bounds__(kThreads) void shapelet_kernel(
    const float* __restrict__ x,
    const float* __restrict__ w,
    const float* __restrict__ pcm,
    float* __restrict__ out)
{
  __shared__ float xs[kT];            // one (b,c) time series row
  __shared__ float ws[kN * kWpad];    // weights for channel c, padded
  __shared__ float pc[kM + 3];        // penalty row for channel c
  __shared__ float red_d[kThreads];
  __shared__ float red_p[kThreads];

  const int tid = threadIdx.x;
  const int b = blockIdx.x >> 3;
  const int c = blockIdx.x & 7;

  const float* gx = x + (b * kC + c) * kT;

  // ---- async stage: x row (1 f32/thread) ----
  async_cp_b32(gx + tid, &xs[tid]);

  // ---- async stage: weights, re-padded to stride 25 (3 f32/thread) ----
  for (int i = tid; i < kN * kL; i += kThreads) {
    const int n = i / kL;
    const int l = i - n * kL;
    async_cp_b32(w + (n * kC + c) * kL + l, &ws[n * kWpad + l]);
  }

  // ---- async stage: penalty row ----
  if (tid < kM)
    async_cp_b32(pcm + c * kM + tid, &pc[tid]);

  wait_async0();
  __syncthreads();

  // lanes 0..31 of a wave hold consecutive n at the SAME m-stripe:
  //  -> xs[] reads are uniform per wave (LDS broadcast)
  //  -> ws[] reads stride 25 (odd) -> no bank conflicts
  const int n  = tid & 63;
  const int mb = tid >> 6;   // 0..7, uniform within a wave

  float wr[kL];
  #pragma unroll
  for (int l = 0; l < kL; ++l) wr[l] = ws[n * kWpad + l];

  float dmin = INFINITY;
  float pmax = -INFINITY;

  for (int m = mb; m < kM; m += 8) {
    const int base = m * kStride;
    float s = 0.0f;
    #pragma unroll
    for (int l = 0; l < kL; ++l)
      s += fabsf(xs[base + l] - wr[l]);
    const float d = pc[m] * (s * (1.0f / kL));
    const float p = __expf(-d * d);
    dmin = fminf(dmin, d);
    pmax = fmaxf(pmax, p);
  }

  red_d[tid] = dmin;
  red_p[tid] = pmax;
  __syncthreads();

  if (tid < kN) {
    float dm = red_d[tid];
    float pm = red_p[tid];
    #pragma unroll
    for (int k = 1; k < 8; ++k) {
      dm = fminf(dm, red_d[k * 64 + tid]);
      pm = fmaxf(pm, red_p[k * 64 + tid]);
    }
    const int o = b * (kN * kC) + tid * kC + c;   // (b, n, c) flattened
    out[o] = pm;                                   // max_p
    out[kB * kN * kC + o] = dm;                    // d_min
  }
}

extern "C" void kernel_launch(void* const* d_in, const int* in_sizes, int n_in,
                              void* d_out, int out_size, void* d_ws, size_t ws_size,
                              hipStream_t stream) {
  const float* x   = (const float*)d_in[0];   // (16, 8, 512)
  const float* w   = (const float*)d_in[1];   // (64, 8, 24)
  const float* pcm = (const float*)d_in[2];   // (8, 245)
  float* out = (float*)d_out;                 // 2 * 16*64*8 = 16384 f32

  shapelet_kernel<<<dim3(kB * kC), dim3(kThreads), 0, stream>>>(x, w, pcm, out);
}